// HierDDLpmC_1382979469617
// MI455X (gfx1250) — compile-verified
//
#include <hip/hip_runtime.h>
#include <hip/hip_bf16.h>

// Problem constants (match reference setup_inputs()).
#define NROWS   2048      // BATCH*SEQ
#define DIM     256       // IN_DIM == OUT_DIM
#define SEQLEN  1024
#define NBATCH  2
#define OUTSEQ  512       // SEQ / RED
#define MAXOUT  512       // Linker second dim
#define LN_EPSF 1e-5f

typedef __attribute__((ext_vector_type(2))) float v2f;
typedef __attribute__((ext_vector_type(8))) float v8f;

// ---------------------------------------------------------------------------
// Kernel A: Y = x @ M^T and R = x @ resW^T via V_WMMA_F32_16X16X4_F32,
// then Z = LayerNorm(Y) in LDS. One workgroup = 16 rows x 256 cols.
// 4 waves; each wave owns 4 column tiles of 16. One A fragment feeds
// 8 WMMAs (4 tiles x 2 weight matrices); all B fragments are fetched
// before the WMMA burst so loads can overlap matrix issue.
// ---------------------------------------------------------------------------
__global__ __launch_bounds__(128) void gemm_ln_kernel(
    const float* __restrict__ x, const float* __restrict__ M,
    const float* __restrict__ resW, const float* __restrict__ gamma,
    const float* __restrict__ beta, float* __restrict__ Z,
    float* __restrict__ R)
{
  __shared__ float sY[16][DIM + 1];
  __shared__ float sMu[16];
  __shared__ float sRs[16];

  const int tid  = threadIdx.x;
  const int wave = tid >> 5;
  const int lane = tid & 31;
  const int half = lane >> 4;   // 0: K0/K1 half, 1: K2/K3 half
  const int l16  = lane & 15;   // M (A) or N (B/C) index
  const int row0 = blockIdx.x * 16;
  const int cbase = wave * 64;

  v8f accY[4], accR[4];
  for (int t = 0; t < 4; ++t)
    for (int r = 0; r < 8; ++r) { accY[t][r] = 0.0f; accR[t][r] = 0.0f; }

  for (int kk = 0; kk < DIM; kk += 4) {
    const int kb = kk + 2 * half;
    // A fragment: 16x4 tile of x. lane<16 -> K={kk,kk+1}; lane>=16 -> K={kk+2,kk+3}
    v2f a;
    const float* xp = x + (row0 + l16) * DIM + kb;
    a.x = xp[0]; a.y = xp[1];

    // Fetch ALL B fragments for this k-step first (Bt[k][n] = W[c0+n][k]).
    v2f bM[4], bR[4];
#pragma unroll
    for (int t = 0; t < 4; ++t) {
      const int c0 = cbase + t * 16;
      const float* mp = M    + (c0 + l16) * DIM + kb;
      const float* rp = resW + (c0 + l16) * DIM + kb;
      bM[t].x = mp[0]; bM[t].y = mp[1];
      bR[t].x = rp[0]; bR[t].y = rp[1];
    }
#pragma unroll
    for (int t = 0; t < 4; ++t) {
      accY[t] = __builtin_amdgcn_wmma_f32_16x16x4_f32(
          false, a, false, bM[t], (short)0, accY[t], false, false);
      accR[t] = __builtin_amdgcn_wmma_f32_16x16x4_f32(
          false, a, false, bR[t], (short)0, accR[t], false, false);
    }
  }

  // Spill Y to LDS (for LN), R straight to global scratch.
#pragma unroll
  for (int t = 0; t < 4; ++t) {
    const int c0 = cbase + t * 16;
#pragma unroll
    for (int r = 0; r < 8; ++r) {
      const int m = r + 8 * half;              // C layout: VGPR r holds M=r / M=r+8
      sY[m][c0 + l16] = accY[t][r];
      R[(row0 + m) * DIM + c0 + l16] = accR[t][r];
    }
  }
  __syncthreads();

  if (tid < 16) {
    float s = 0.0f;
    for (int j = 0; j < DIM; ++j) s += sY[tid][j];
    const float mu = s * (1.0f / (float)DIM);
    float v = 0.0f;
    for (int j = 0; j < DIM; ++j) {
      const float d = sY[tid][j] - mu;
      v = fmaf(d, d, v);
    }
    sMu[tid] = mu;
    sRs[tid] = rsqrtf(v * (1.0f / (float)DIM) + LN_EPSF);
  }
  __syncthreads();

#pragma unroll
  for (int e = 0; e < 32; ++e) {
    const int flat = e * 128 + tid;            // coalesced: consecutive tid -> cols
    const int m = flat >> 8;
    const int j = flat & (DIM - 1);
    Z[(row0 + m) * DIM + j] =
        (sY[m][j] - sMu[m]) * sRs[m] * gamma[j] + beta[j];
  }
}

// ---------------------------------------------------------------------------
// Kernel B: T[k,i] = sum_j Z[k,j] * P[i,j] * cos(2*pi*k / (i*256 + j + 2)).
// v_cos_f32 takes its input in revolutions (period 1.0), so
// cos(2*pi*k/p) == v_cos(fract(k * rcp(p))): 1 hw cos per term, rcp amortized
// over KB=8 k-rows (the period depends only on (i,j)).
// One workgroup = 8 k-rows x all 256 i; thread = one i.
// ---------------------------------------------------------------------------
#define KB 8
__global__ __launch_bounds__(256) void phi_kernel(
    const float* __restrict__ Z, const float* __restrict__ P,
    float* __restrict__ T)
{
  __shared__ float sZ[KB][DIM];
  const int tid = threadIdx.x;
  const int k0  = blockIdx.x * KB;

#pragma unroll
  for (int e = 0; e < KB; ++e)
    sZ[e][tid] = Z[(k0 + e) * DIM + tid];
  __syncthreads();

  float acc[KB];
#pragma unroll
  for (int e = 0; e < KB; ++e) acc[e] = 0.0f;

  const int i = tid;
  const float* Prow = P + i * DIM;
  const float pbase = (float)(i * DIM + 2);    // exact in f32 (< 2^24)

#pragma unroll 2
  for (int j = 0; j < DIM; ++j) {
    const float pf   = pbase + (float)j;
    const float rinv = __builtin_amdgcn_rcpf(pf);   // amortized over KB rows
    const float pw   = Prow[j];
#pragma unroll
    for (int e = 0; e < KB; ++e) {
      const float kf = (float)(k0 + e);
      const float q  = __builtin_amdgcn_fractf(kf * rinv);
      const float c  = __builtin_amdgcn_cosf(q);    // cos(2*pi*q)
      acc[e] = fmaf(sZ[e][j] * pw, c, acc[e]);
    }
  }

#pragma unroll
  for (int e = 0; e < KB; ++e)
    T[(k0 + e) * DIM + i] = acc[e];               // coalesced across i
}

// ---------------------------------------------------------------------------
// Kernel C: out[b*512+t, d] = sum_s L[s,t]*T[b,s,d] + sum_s rL[s,t]*R[b,s,d].
// 2x2 register tile per wave: A fragments (L, rL) are reused across 2 d-tiles
// and B fragments (T, R) across 2 t-tiles -> 16 loads per 8 WMMAs per k-step
// (vs 8 loads / 2 WMMAs untiled). Both products accumulate into the SAME
// accumulators since the output is their sum.
// ---------------------------------------------------------------------------
__global__ __launch_bounds__(128) void final_gemm_kernel(
    const float* __restrict__ Lk, const float* __restrict__ rL,
    const float* __restrict__ T, const float* __restrict__ R,
    float* __restrict__ out)
{
  const int tid  = threadIdx.x;
  const int wave = tid >> 5;
  const int lane = tid & 31;
  const int half = lane >> 4;
  const int l16  = lane & 15;

  const int job = blockIdx.x * 4 + wave;         // 0..255
  const int b   = job >> 7;                      // 128 jobs per batch
  const int rem = job & 127;
  const int t0  = (rem >> 3) * 32;               // 16 t-pairs of 32 rows
  const int d0  = (rem & 7) * 32;                // 8 d-pairs of 32 cols

  v8f acc[2][2];
  for (int u = 0; u < 2; ++u)
    for (int v = 0; v < 2; ++v)
      for (int r = 0; r < 8; ++r) acc[u][v][r] = 0.0f;

  const float* Tb = T + b * SEQLEN * DIM;
  const float* Rb = R + b * SEQLEN * DIM;

  for (int kk = 0; kk < SEQLEN; kk += 4) {
    const int krow = kk + 2 * half;
    v2f aL[2], aR[2], bT[2], bR[2];
#pragma unroll
    for (int u = 0; u < 2; ++u) {
      // A[m][k] = L[k][t0+m] (transposed access, coalesced over lanes)
      const int tt = t0 + u * 16 + l16;
      aL[u].x = Lk[(krow + 0) * MAXOUT + tt];
      aL[u].y = Lk[(krow + 1) * MAXOUT + tt];
      aR[u].x = rL[(krow + 0) * MAXOUT + tt];
      aR[u].y = rL[(krow + 1) * MAXOUT + tt];
      // B[k][n] = T[b, k, d0+n]
      const int dd = d0 + u * 16 + l16;
      bT[u].x = Tb[(krow + 0) * DIM + dd];
      bT[u].y = Tb[(krow + 1) * DIM + dd];
      bR[u].x = Rb[(krow + 0) * DIM + dd];
      bR[u].y = Rb[(krow + 1) * DIM + dd];
    }
#pragma unroll
    for (int u = 0; u < 2; ++u)
#pragma unroll
      for (int v = 0; v < 2; ++v) {
        acc[u][v] = __builtin_amdgcn_wmma_f32_16x16x4_f32(
            false, aL[u], false, bT[v], (short)0, acc[u][v], false, false);
        acc[u][v] = __builtin_amdgcn_wmma_f32_16x16x4_f32(
            false, aR[u], false, bR[v], (short)0, acc[u][v], false, false);
      }
  }

#pragma unroll
  for (int u = 0; u < 2; ++u)
#pragma unroll
    for (int v = 0; v < 2; ++v)
#pragma unroll
      for (int r = 0; r < 8; ++r)
        out[(b * OUTSEQ + t0 + u * 16 + r + 8 * half) * DIM + d0 + v * 16 + l16]
            = acc[u][v][r];
}

// ---------------------------------------------------------------------------
extern "C" void kernel_launch(void* const* d_in, const int* in_sizes, int n_in,
                              void* d_out, int out_size, void* d_ws, size_t ws_size,
                              hipStream_t stream)
{
  const float* x     = (const float*)d_in[0];
  const float* M     = (const float*)d_in[1];
  const float* P     = (const float*)d_in[2];
  const float* Lk    = (const float*)d_in[3];
  const float* gamma = (const float*)d_in[4];
  const float* beta  = (const float*)d_in[5];
  const float* resW  = (const float*)d_in[6];
  const float* rL    = (const float*)d_in[7];
  // d_in[8] = actual_seq_len (== SEQ, static in this problem)

  float* Z = (float*)d_ws;                 // 2048*256 f32 = 2 MB
  float* R = Z + (size_t)NROWS * DIM;      // 2 MB
  float* T = R + (size_t)NROWS * DIM;      // 2 MB
  float* out = (float*)d_out;

  gemm_ln_kernel<<<NROWS / 16, 128, 0, stream>>>(x, M, resW, gamma, beta, Z, R);
  phi_kernel<<<NROWS / KB, 256, 0, stream>>>(Z, P, T);
  final_gemm_kernel<<<64, 128, 0, stream>>>(Lk, rL, T, R, out);
}